// SceneNet_80376017977637
// MI455X (gfx1250) — compile-verified
//
#include <hip/hip_runtime.h>
#include <math.h>

#define HH 128
#define WW 128
#define NPIX (HH*WW)        // 16384
#define NEDGE (25*NPIX)     // 409600
#define BB 2
#define QQ 128
#define MM 16
#define NITERS 32
#define EPSF 1e-8f

typedef __attribute__((ext_vector_type(2))) float v2f;
typedef __attribute__((ext_vector_type(8))) float v8f;

// D = A(16x4 f32) * B(4x16 f32) + C(16x16 f32)
// A layout: lane L holds row M=L%16; VGPR0 = K{0|2}, VGPR1 = K{1|3} (halves of wave)
// C/D layout: VGPR r -> M = r (lanes 0-15) / r+8 (lanes 16-31), N = lane%16
static __device__ __forceinline__ v8f wmma4(v2f a, v2f b, v8f c) {
  return __builtin_amdgcn_wmma_f32_16x16x4_f32(false, a, false, b, (short)0, c,
                                               false, false);
}

// ---------------- conv1: 3x3, C=3 -> D=32, relu (VALU; tiny FLOPs) ----------
__global__ void k_conv1(const float* __restrict__ x, const float* __restrict__ w1,
                        const float* __restrict__ b1, float* __restrict__ f1) {
  __shared__ float sw[9 * 3 * 32];
  __shared__ float sb[32];
  int tid = threadIdx.x;
  for (int v = tid; v < 864; v += 128) sw[v] = w1[v];
  if (tid < 32) sb[tid] = b1[tid];
  __syncthreads();

  int pix = blockIdx.x * 128 + tid;          // 0 .. B*N-1
  int b = pix / NPIX, p = pix % NPIX;
  int i = p / WW, j = p % WW;
  float acc[32];
#pragma unroll
  for (int d = 0; d < 32; ++d) acc[d] = sb[d];
  for (int ki = 0; ki < 3; ++ki) {
    int gi = i + ki - 1;
    if (gi < 0 || gi >= HH) continue;
    for (int kj = 0; kj < 3; ++kj) {
      int gj = j + kj - 1;
      if (gj < 0 || gj >= WW) continue;
      const float* xp = &x[(((size_t)b * HH + gi) * WW + gj) * 3];
      const float* wp = &sw[(ki * 3 + kj) * 3 * 32];
#pragma unroll
      for (int c = 0; c < 3; ++c) {
        float xv = xp[c];
#pragma unroll
        for (int d = 0; d < 32; ++d) acc[d] = fmaf(xv, wp[c * 32 + d], acc[d]);
      }
    }
  }
  float* op = &f1[((size_t)b * NPIX + p) * 32];
#pragma unroll
  for (int d = 0; d < 32; ++d) op[d] = fmaxf(acc[d], 0.0f);
}

// ---------------- conv2: 3x3, 32->32, relu via WMMA implicit GEMM -----------
static __device__ __forceinline__ float loadA2(const float* __restrict__ f1,
                                               int b, int i, int j, int k) {
  // k = (ki*3+kj)*32 + c  (matches HWIO layout of w_conv2 flattened)
  int tap = k >> 5, c = k & 31;
  int ki = tap / 3, kj = tap % 3;
  int gi = i + ki - 1, gj = j + kj - 1;
  bool ok = (gi >= 0) & (gi < HH) & (gj >= 0) & (gj < WW);
  size_t idx = ok ? (((size_t)b * NPIX + (size_t)gi * WW + gj) * 32 + c) : 0;
  float v = f1[idx];
  return ok ? v : 0.0f;
}

__global__ void k_conv2(const float* __restrict__ f1, const float* __restrict__ w2,
                        const float* __restrict__ b2, float* __restrict__ f2) {
  int tile = blockIdx.x;                       // B*H*(W/16) tiles, 1 wave each
  int b   = tile / (HH * (WW / 16));
  int rem = tile % (HH * (WW / 16));
  int i   = rem / (WW / 16);
  int j0  = (rem % (WW / 16)) * 16;
  int lane = threadIdx.x;
  int l = lane & 15, half = lane >> 4;
  int j = j0 + l;                              // A row = pixel column

  v8f c0 = {}, c1 = {};
  for (int kk = 0; kk < 72; ++kk) {            // K = 288 = 72*4
    int k0 = kk * 4 + half * 2;
    v2f a;
    a.x = loadA2(f1, b, i, j, k0);
    a.y = loadA2(f1, b, i, j, k0 + 1);
    v2f bw0, bw1;
    bw0.x = w2[k0 * 32 + l];        bw0.y = w2[(k0 + 1) * 32 + l];
    bw1.x = w2[k0 * 32 + 16 + l];   bw1.y = w2[(k0 + 1) * 32 + 16 + l];
    c0 = wmma4(a, bw0, c0);
    c1 = wmma4(a, bw1, c1);
  }
#pragma unroll
  for (int r = 0; r < 8; ++r) {
    int pj = j0 + r + half * 8;
    float* op = &f2[(((size_t)b * NPIX) + (size_t)i * WW + pj) * 32];
    op[l]      = fmaxf(c0[r] + b2[l], 0.0f);
    op[16 + l] = fmaxf(c1[r] + b2[16 + l], 0.0f);
  }
}

// ---------------- K / Q 1x1 projections via WMMA (K=32) ---------------------
__global__ void k_kq(const float* __restrict__ f2, const float* __restrict__ wk,
                     const float* __restrict__ bk, const float* __restrict__ wq,
                     float* __restrict__ ks, float* __restrict__ qs) {
  int tile = blockIdx.x;                       // B*N/16 tiles, 1 wave each
  int b  = tile / (NPIX / 16);
  int p0 = (tile % (NPIX / 16)) * 16;
  int lane = threadIdx.x;
  int l = lane & 15, half = lane >> 4;
  const float* ap = &f2[((size_t)b * NPIX + p0 + l) * 32];

  v8f ck0 = {}, ck1 = {}, cq0 = {}, cq1 = {};
  for (int kk = 0; kk < 8; ++kk) {
    int k0 = kk * 4 + half * 2;
    v2f a; a.x = ap[k0]; a.y = ap[k0 + 1];
    v2f bk0v, bk1v, bq0v, bq1v;
    bk0v.x = wk[k0 * 32 + l];       bk0v.y = wk[(k0 + 1) * 32 + l];
    bk1v.x = wk[k0 * 32 + 16 + l];  bk1v.y = wk[(k0 + 1) * 32 + 16 + l];
    bq0v.x = wq[k0 * 32 + l];       bq0v.y = wq[(k0 + 1) * 32 + l];
    bq1v.x = wq[k0 * 32 + 16 + l];  bq1v.y = wq[(k0 + 1) * 32 + 16 + l];
    ck0 = wmma4(a, bk0v, ck0);
    ck1 = wmma4(a, bk1v, ck1);
    cq0 = wmma4(a, bq0v, cq0);
    cq1 = wmma4(a, bq1v, cq1);
  }
#pragma unroll
  for (int r = 0; r < 8; ++r) {
    int p = p0 + r + half * 8;
    float* kp = &ks[((size_t)b * NPIX + p) * 32];
    float* qp = &qs[((size_t)b * NPIX + p) * 32];
    kp[l]      = ck0[r] + bk[l];
    kp[16 + l] = ck1[r] + bk[16 + l];
    qp[l]      = cq0[r];
    qp[16 + l] = cq1[r];
  }
}

// --------- edge attention: exp(cos_sim - 1) + deterministic partial sums ----
__global__ void k_attn(const float* __restrict__ qs, const float* __restrict__ ks,
                       const int* __restrict__ edges, float* __restrict__ wexp,
                       float* __restrict__ part) {
  __shared__ float red[256];
  int b = blockIdx.y;
  int e = blockIdx.x * 256 + threadIdx.x;
  float w = 0.0f;
  if (e < NEDGE) {
    int r = edges[e], c = edges[NEDGE + e];
    const float* qp = &qs[((size_t)b * NPIX + r) * 32];
    const float* kp = &ks[((size_t)b * NPIX + c) * 32];
    float num = 0.f, qq = 0.f, kk = 0.f;
#pragma unroll
    for (int d = 0; d < 32; ++d) {
      float qv = qp[d], kv = kp[d];
      num = fmaf(qv, kv, num);
      qq  = fmaf(qv, qv, qq);
      kk  = fmaf(kv, kv, kk);
    }
    float den = fmaxf(sqrtf(qq) * sqrtf(kk), EPSF);
    w = __expf(num / den - 1.0f);   // |cos| <= 1 => safe fixed max
    wexp[(size_t)b * NEDGE + e] = w;
  }
  red[threadIdx.x] = w;
  __syncthreads();
  for (int s = 128; s > 0; s >>= 1) {
    if (threadIdx.x < s) red[threadIdx.x] += red[threadIdx.x + s];
    __syncthreads();
  }
  if (threadIdx.x == 0) part[(size_t)b * gridDim.x + blockIdx.x] = red[0];
}

__global__ void k_sum(const float* __restrict__ part, float* __restrict__ sums,
                      int nPart) {
  __shared__ float red[256];
  int b = blockIdx.x;
  float s = 0.f;
  for (int i = threadIdx.x; i < nPart; i += 256) s += part[(size_t)b * nPart + i];
  red[threadIdx.x] = s;
  __syncthreads();
  for (int st = 128; st > 0; st >>= 1) {
    if (threadIdx.x < st) red[threadIdx.x] += red[threadIdx.x + st];
    __syncthreads();
  }
  if (threadIdx.x == 0) sums[b] = red[0];
}

__global__ void k_normw(float* __restrict__ wexp, const float* __restrict__ sums) {
  size_t idx = (size_t)blockIdx.x * 256 + threadIdx.x;  // B*NEDGE exact
  int b = (int)(idx / NEDGE);
  wexp[idx] = wexp[idx] / sums[b];
}

// ------- propagation: 5x5 weighted gather + L2 normalize, LDS-tiled ---------
// Halo staging uses CDNA5 async global->LDS copies (ASYNCcnt path, no VGPR
// round trip); one s_wait_asynccnt 0 before the workgroup barrier.
__global__ void k_prop(const float* __restrict__ hsrc, float* __restrict__ hdst,
                       const float* __restrict__ w) {
  extern __shared__ float lds[];               // 12*12 halo x 128 ch = 72 KB
  int b  = blockIdx.z;
  int i0 = blockIdx.y * 8, j0 = blockIdx.x * 8;
  int tid = threadIdx.x;

  for (int v = tid; v < 144 * 32; v += 256) {  // float4 granularity
    int hcell = v >> 5, c4 = v & 31;
    int li = hcell / 12, lj = hcell % 12;
    int gi = i0 - 2 + li; gi = gi < 0 ? 0 : (gi > HH - 1 ? HH - 1 : gi);
    int gj = j0 - 2 + lj; gj = gj < 0 ? 0 : (gj > WW - 1 ? WW - 1 : gj);
    const float4* sp =
        (const float4*)&hsrc[((size_t)b * NPIX + (size_t)gi * WW + gj) * QQ] + c4;
#if defined(__HIP_DEVICE_COMPILE__) && defined(__gfx1250__)
    // LDS aperture: flat addr[31:0] == LDS byte offset (ISA 10.2)
    unsigned lA = (unsigned)(uintptr_t)(&lds[(hcell * 32 + c4) * 4]);
    asm volatile("global_load_async_to_lds_b128 %0, %1, off"
                 :: "v"(lA), "v"(sp) : "memory");
#else
    ((float4*)lds)[hcell * 32 + c4] = *sp;
#endif
  }
#if defined(__HIP_DEVICE_COMPILE__) && defined(__gfx1250__)
  asm volatile("s_wait_asynccnt 0x0" ::: "memory");
#endif
  __syncthreads();

  int wave = tid >> 5, lane = tid & 31;        // lane -> channels 4l..4l+3
  for (int pp = 0; pp < 8; ++pp) {
    int pi = wave, pj = pp;
    int gp = (i0 + pi) * WW + (j0 + pj);
    float ax = 0.f, ay = 0.f, az = 0.f, aw = 0.f;
#pragma unroll
    for (int o = 0; o < 25; ++o) {             // o = (di+2)*5 + (dj+2)
      float wv = w[(size_t)b * NEDGE + (size_t)o * NPIX + gp];
      float4 hv = ((const float4*)lds)[((pi + o / 5) * 12 + (pj + o % 5)) * 32 + lane];
      ax = fmaf(wv, hv.x, ax); ay = fmaf(wv, hv.y, ay);
      az = fmaf(wv, hv.z, az); aw = fmaf(wv, hv.w, aw);
    }
    float ss = ax * ax + ay * ay + az * az + aw * aw;
#pragma unroll
    for (int m = 16; m >= 1; m >>= 1) ss += __shfl_xor(ss, m, 32);
    float sc = 1.0f / (sqrtf(ss) + EPSF);
    float4 o4; o4.x = ax * sc; o4.y = ay * sc; o4.z = az * sc; o4.w = aw * sc;
    ((float4*)&hdst[((size_t)b * NPIX + gp) * QQ])[lane] = o4;
  }
}

// ------------- mask head: h @ w_mask (WMMA, K=128) + softmax ----------------
__global__ void k_mask(const float* __restrict__ h, const float* __restrict__ wm,
                       float* __restrict__ out) {
  __shared__ float tl[16 * 16];
  int tile = blockIdx.x;                       // B*N/16 tiles, 1 wave each
  int b  = tile / (NPIX / 16);
  int p0 = (tile % (NPIX / 16)) * 16;
  int lane = threadIdx.x;
  int l = lane & 15, half = lane >> 4;
  const float* ap = &h[((size_t)b * NPIX + p0 + l) * QQ];

  v8f c = {};
  for (int kk = 0; kk < 32; ++kk) {
    int k0 = kk * 4 + half * 2;
    v2f a; a.x = ap[k0]; a.y = ap[k0 + 1];
    v2f bv; bv.x = wm[k0 * 16 + l]; bv.y = wm[(k0 + 1) * 16 + l];
    c = wmma4(a, bv, c);
  }
#pragma unroll
  for (int r = 0; r < 8; ++r) tl[(r + half * 8) * 16 + l] = c[r];
  __syncthreads();

  if (lane < 16) {
    int p = p0 + lane;
    float mx = -1e30f;
#pragma unroll
    for (int m = 0; m < 16; ++m) mx = fmaxf(mx, tl[lane * 16 + m]);
    float ev[16]; float s = 0.f;
#pragma unroll
    for (int m = 0; m < 16; ++m) { ev[m] = __expf(tl[lane * 16 + m] - mx); s += ev[m]; }
    float inv = 1.0f / s;
#pragma unroll
    for (int m = 0; m < 16; ++m)
      out[((size_t)b * MM + m) * NPIX + p] = ev[m] * inv;   // (B,M,H,W)
  }
}

// ----------------------------------------------------------------------------
extern "C" void kernel_launch(void* const* d_in, const int* in_sizes, int n_in,
                              void* d_out, int out_size, void* d_ws, size_t ws_size,
                              hipStream_t stream) {
  (void)in_sizes; (void)n_in; (void)out_size; (void)ws_size;
  const float* x    = (const float*)d_in[0];
  const int*   edge = (const int*)d_in[1];
  const float* w1   = (const float*)d_in[2];
  const float* b1   = (const float*)d_in[3];
  const float* w2   = (const float*)d_in[4];
  const float* b2   = (const float*)d_in[5];
  const float* wk   = (const float*)d_in[6];
  const float* bk   = (const float*)d_in[7];
  const float* wq   = (const float*)d_in[8];
  const float* h0i  = (const float*)d_in[9];
  const float* wm   = (const float*)d_in[10];
  float* out = (float*)d_out;

  float* ws = (float*)d_ws;
  float* feat1 = ws;  ws += (size_t)BB * NPIX * 32;
  float* feat2 = ws;  ws += (size_t)BB * NPIX * 32;
  float* ksb   = ws;  ws += (size_t)BB * NPIX * 32;
  float* qsb   = ws;  ws += (size_t)BB * NPIX * 32;
  float* wexp  = ws;  ws += (size_t)BB * NEDGE;
  float* part  = ws;  ws += (size_t)BB * (NEDGE / 256);
  float* sums  = ws;  ws += 8;
  float* hA    = ws;  ws += (size_t)BB * NPIX * QQ;
  float* hB    = ws;

  k_conv1<<<BB * NPIX / 128, 128, 0, stream>>>(x, w1, b1, feat1);
  k_conv2<<<BB * HH * (WW / 16), 32, 0, stream>>>(feat1, w2, b2, feat2);
  k_kq<<<BB * NPIX / 16, 32, 0, stream>>>(feat2, wk, bk, wq, ksb, qsb);
  k_attn<<<dim3(NEDGE / 256, BB), 256, 0, stream>>>(qsb, ksb, edge, wexp, part);
  k_sum<<<BB, 256, 0, stream>>>(part, sums, NEDGE / 256);
  k_normw<<<BB * NEDGE / 256, 256, 0, stream>>>(wexp, sums);

  const float* src = h0i;                      // never mutated
  float* bufs[2] = { hA, hB };
  for (int it = 0; it < NITERS; ++it) {
    float* dst = bufs[it & 1];
    k_prop<<<dim3(WW / 8, HH / 8, BB), 256, 144 * QQ * sizeof(float), stream>>>(
        src, dst, wexp);
    src = dst;
  }
  k_mask<<<BB * NPIX / 16, 32, 0, stream>>>(src, wm, out);
}